// DeformableTransformer_11175504904382
// MI455X (gfx1250) — compile-verified
//
#include <hip/hip_runtime.h>
#include <hip/hip_bf16.h>

// ---------------------------------------------------------------------------
// Deformable transformer encoder layer for MI455X (gfx1250, wave32).
//
// Roofline: ~40 GFLOP fp32 of GEMM + ~300 MB activation traffic. At
// 23.3 TB/s HBM this layer is bandwidth/latency bound (~13 us of traffic),
// so we keep full fp32 precision and use V_WMMA_F32_16X16X4_F32 for every
// GEMM. The 27 MB `value` tensor fits entirely in the 192 MB L2, so the
// random bilinear gathers are L2-resident.
//
// GEMM shape: block = 256 threads (8 wave32), block tile 64(M) x 128(N).
// Each wave owns 4 M-subtiles sharing one B fragment per WMMA group -> 4
// independent v_wmma per B load (hides the WMMA->WMMA RAW hazard, cuts B
// traffic 4x). A tile staged to LDS via GLOBAL_LOAD_ASYNC_TO_LDS_B128
// (ASYNCcnt path; builtin confirmed present on this toolchain), else float4.
// ---------------------------------------------------------------------------

typedef float v2f __attribute__((ext_vector_type(2)));
typedef float v8f __attribute__((ext_vector_type(8)));
// Matches the builtin's parameter type exactly as printed by clang:
//   __attribute__((__vector_size__(4 * sizeof(int)))) int __device__ *
typedef int v4i_vs __attribute__((vector_size(4 * sizeof(int))));

#define AS1 __attribute__((address_space(1)))
#define AS3 __attribute__((address_space(3)))

#if defined(__has_builtin)
#if __has_builtin(__builtin_amdgcn_global_load_async_to_lds_b128)
#define HAVE_ASYNC_LDS 1
#endif
#endif
#ifndef HAVE_ASYNC_LDS
#define HAVE_ASYNC_LDS 0
#endif

namespace {
constexpr int Bc  = 2;
constexpr int Sc  = 13294;           // 100^2 + 50^2 + 25^2 + 13^2
constexpr int Mc  = Bc * Sc;         // 26588 rows
constexpr int NHc = 8;
}

// ---------------------------------------------------------------------------
// q = src + pos (elementwise)
// ---------------------------------------------------------------------------
__global__ void add_kernel(float* __restrict__ o, const float* __restrict__ a,
                           const float* __restrict__ b, int n) {
    int i = blockIdx.x * blockDim.x + threadIdx.x;
    if (i < n) o[i] = a[i] + b[i];
}

// ---------------------------------------------------------------------------
// WMMA fp32 GEMM: C[M,N] = A[M,K] @ W[K,N] + bias, optional ReLU.
//
// Fragment layouts (CDNA5 ISA, f32 16x16x4):
//   A: lane m = lane&15; VGPRs {K0,K1} for lanes 0-15, {K2,K3} for lanes 16-31
//   B: row K striped across lanes (N = lane&15), same K half-split
//   C/D: VGPR r -> M = r + 8*(lane>=16), N = lane&15
//
// LDS tile 64 x 132 floats (pad 132 % 64banks = 4 -> conflict-free fragment
// reads for all four 16-row subtiles). N must be a multiple of 128 (128 /
// 256 / 1024 here); M tail rows zero-filled in LDS, stores uniform-guarded.
// ---------------------------------------------------------------------------
template <int EPI> // 0 = none, 1 = relu
__global__ __launch_bounds__(256) void gemm_wmma_kernel(
    const float* __restrict__ A, const float* __restrict__ W,
    const float* __restrict__ bias, float* __restrict__ C,
    int Mr, int N, int K) {
    __shared__ float lA[64][132];

    const int tid   = threadIdx.x;
    const int lane  = tid & 31;
    const int wave  = tid >> 5;
    const int half  = lane >> 4;   // 0: lanes 0-15, 1: lanes 16-31
    const int lm    = lane & 15;
    const int mbase = blockIdx.y * 64;
    const int n     = blockIdx.x * 128 + wave * 16 + lm;

    v8f acc0 = {}, acc1 = {}, acc2 = {}, acc3 = {};

    for (int k0 = 0; k0 < K; k0 += 128) {
        // ---- Stage A[mbase:mbase+64, k0:k0+128] into LDS (2048 float4) ----
#pragma unroll
        for (int i = 0; i < 8; ++i) {
            const int fi  = tid + i * 256;     // float4 index: 0..2047
            const int r   = fi >> 5;           // 0..63
            const int c4  = (fi & 31) * 4;     // 0..124
            const int row = mbase + r;
            if (row < Mr) {
                const float* gp = A + (size_t)row * K + k0 + c4;
#if HAVE_ASYNC_LDS
                // CDNA5 async copy: global -> LDS without touching VGPRs.
                __builtin_amdgcn_global_load_async_to_lds_b128(
                    (AS1 v4i_vs*)(unsigned long long)gp,
                    (AS3 v4i_vs*)(unsigned int)(unsigned long long)&lA[r][c4],
                    0, 0);
#else
                *(float4*)&lA[r][c4] = *(const float4*)gp;
#endif
                if (k0 + 128 < K)
                    __builtin_prefetch(gp + 128, 0, 1);
            } else {
                lA[r][c4 + 0] = 0.0f;
                lA[r][c4 + 1] = 0.0f;
                lA[r][c4 + 2] = 0.0f;
                lA[r][c4 + 3] = 0.0f;
            }
        }
#if HAVE_ASYNC_LDS
#if __has_builtin(__builtin_amdgcn_s_wait_asynccnt)
        __builtin_amdgcn_s_wait_asynccnt(0);
#else
        asm volatile("s_wait_asynccnt 0x0" ::: "memory");
#endif
#endif
        __syncthreads();

        // ---- 32 WMMA groups; each B fragment feeds 4 independent WMMAs ----
        const float* Wp = W + (size_t)k0 * N + n;
#pragma unroll 4
        for (int kk = 0; kk < 128; kk += 4) {
            const int ko = kk + half * 2;
            v2f bv;
            bv.x = Wp[ko * N];
            bv.y = Wp[(ko + 1) * N];
            v2f a0, a1, a2, a3;
            a0.x = lA[lm][ko];      a0.y = lA[lm][ko + 1];
            a1.x = lA[16 + lm][ko]; a1.y = lA[16 + lm][ko + 1];
            a2.x = lA[32 + lm][ko]; a2.y = lA[32 + lm][ko + 1];
            a3.x = lA[48 + lm][ko]; a3.y = lA[48 + lm][ko + 1];
            acc0 = __builtin_amdgcn_wmma_f32_16x16x4_f32(
                false, a0, false, bv, (short)0, acc0, false, false);
            acc1 = __builtin_amdgcn_wmma_f32_16x16x4_f32(
                false, a1, false, bv, (short)0, acc1, false, false);
            acc2 = __builtin_amdgcn_wmma_f32_16x16x4_f32(
                false, a2, false, bv, (short)0, acc2, false, false);
            acc3 = __builtin_amdgcn_wmma_f32_16x16x4_f32(
                false, a3, false, bv, (short)0, acc3, false, false);
        }
        __syncthreads();
    }

    // ---- Epilogue: bias (+ReLU), int32 addressing, uniform tail guard ----
    const float bn = bias[n];
    if (mbase + 64 <= Mr) {
#pragma unroll
        for (int ms = 0; ms < 4; ++ms) {
            const v8f a = (ms == 0) ? acc0 : (ms == 1) ? acc1
                        : (ms == 2) ? acc2 : acc3;
            const int mb = mbase + ms * 16 + half * 8;
#pragma unroll
            for (int r = 0; r < 8; ++r) {
                float v = a[r] + bn;
                if (EPI == 1) v = fmaxf(v, 0.0f);
                C[(mb + r) * N + n] = v;
            }
        }
    } else {
#pragma unroll
        for (int ms = 0; ms < 4; ++ms) {
            const v8f a = (ms == 0) ? acc0 : (ms == 1) ? acc1
                        : (ms == 2) ? acc2 : acc3;
            const int mb = mbase + ms * 16 + half * 8;
#pragma unroll
            for (int r = 0; r < 8; ++r) {
                const int m = mb + r;
                float v = a[r] + bn;
                if (EPI == 1) v = fmaxf(v, 0.0f);
                if (m < Mr) C[m * N + n] = v;
            }
        }
    }
}

// ---------------------------------------------------------------------------
// Softmax over the last 16 (NL*NP) values per (b,s,head). One thread per row
// of 16 (rows are contiguous: attn buffer is (b*S+s, NH*16)).
// ---------------------------------------------------------------------------
__global__ void softmax16_kernel(float* __restrict__ attn, int nrows) {
    const int t = blockIdx.x * blockDim.x + threadIdx.x;
    if (t >= nrows) return;
    float* row = attn + (size_t)t * 16;
    float m = row[0];
#pragma unroll
    for (int i = 1; i < 16; ++i) m = fmaxf(m, row[i]);
    float e[16];
    float s = 0.0f;
#pragma unroll
    for (int i = 0; i < 16; ++i) { e[i] = __expf(row[i] - m); s += e[i]; }
    const float inv = 1.0f / s;
#pragma unroll
    for (int i = 0; i < 16; ++i) row[i] = e[i] * inv;
}

// ---------------------------------------------------------------------------
// Deformable attention sampling. One wave32 per (b, s, head); lane = channel
// (DH = 32). All addressing math is wave-uniform; the 32-lane gather of one
// head-slice is a single coalesced 128B load, L2-resident (value = 27 MB).
// acc[b,s,h,dh] = sum_{l,p,corner} attn * bilinear_wgt * value[gather]
// ---------------------------------------------------------------------------
__global__ void deform_attn_kernel(const float* __restrict__ value,
                                   const float* __restrict__ offs,
                                   const float* __restrict__ attn,
                                   const float* __restrict__ refp,
                                   float* __restrict__ acc) {
    const int lane = threadIdx.x & 31;
    const int gw   = blockIdx.x * (blockDim.x >> 5) + (threadIdx.x >> 5);
    if (gw >= Mc * NHc) return;

    const int nh = gw % NHc;
    const int bs = gw / NHc;          // b*S + s
    const int b  = bs / Sc;

    const int HWl[4]   = {100, 50, 25, 13};
    const int basel[4] = {0, 10000, 12500, 13125};

    const float* offrow  = offs + (size_t)bs * 256 + nh * 32;  // (NL,NP,2)
    const float* attnrow = attn + (size_t)bs * 128 + nh * 16;  // (NL,NP)
    const float* refrow  = refp + (size_t)bs * 8;              // (NL,2)

    float a = 0.0f;
#pragma unroll
    for (int l = 0; l < 4; ++l) {
        const int   HW  = HWl[l];
        const float fHW = (float)HW;
        const float rx  = refrow[l * 2 + 0];
        const float ry  = refrow[l * 2 + 1];
        const float* vb = value + ((size_t)(b * Sc + basel[l])) * 256 + nh * 32 + lane;
#pragma unroll
        for (int p = 0; p < 4; ++p) {
            const float ox = offrow[l * 8 + p * 2 + 0];
            const float oy = offrow[l * 8 + p * 2 + 1];
            const float aw = attnrow[l * 4 + p];
            const float lx = rx + ox / fHW;   // offset_norm = [W, H] (square)
            const float ly = ry + oy / fHW;
            const float x  = lx * fHW - 0.5f;
            const float y  = ly * fHW - 0.5f;
            const float x0 = floorf(x);
            const float y0 = floorf(y);
#pragma unroll
            for (int c = 0; c < 4; ++c) {
                const float xi = x0 + (float)(c & 1);
                const float yi = y0 + (float)(c >> 1);
                if (xi >= 0.0f && xi < fHW && yi >= 0.0f && yi < fHW) {
                    const float wgt =
                        (1.0f - fabsf(x - xi)) * (1.0f - fabsf(y - yi)) * aw;
                    const int idx = (int)yi * HW + (int)xi;
                    a += wgt * vb[(size_t)idx * 256];
                }
            }
        }
    }
    acc[(size_t)bs * 256 + nh * 32 + lane] = a;
}

// ---------------------------------------------------------------------------
// out = LayerNorm(a + b) * gamma + beta over D=256. One wave32 per row; each
// lane handles 8 strided channels; mean/var via __shfl_xor wave32 reduction.
// ---------------------------------------------------------------------------
__global__ void residual_ln_kernel(const float* __restrict__ a,
                                   const float* __restrict__ b,
                                   const float* __restrict__ g,
                                   const float* __restrict__ be,
                                   float* __restrict__ out, int Mr) {
    const int lane = threadIdx.x & 31;
    const int row  = blockIdx.x * (blockDim.x >> 5) + (threadIdx.x >> 5);
    if (row >= Mr) return;
    const size_t base = (size_t)row * 256;

    float v[8];
    float s = 0.0f;
#pragma unroll
    for (int i = 0; i < 8; ++i) {
        const int c = lane + 32 * i;
        v[i] = a[base + c] + b[base + c];
        s += v[i];
    }
#pragma unroll
    for (int o = 16; o >= 1; o >>= 1) s += __shfl_xor(s, o, 32);
    const float mean = s * (1.0f / 256.0f);

    float var = 0.0f;
#pragma unroll
    for (int i = 0; i < 8; ++i) {
        const float d = v[i] - mean;
        var += d * d;
    }
#pragma unroll
    for (int o = 16; o >= 1; o >>= 1) var += __shfl_xor(var, o, 32);
    const float inv = rsqrtf(var * (1.0f / 256.0f) + 1e-5f);

#pragma unroll
    for (int i = 0; i < 8; ++i) {
        const int c = lane + 32 * i;
        out[base + c] = (v[i] - mean) * inv * g[c] + be[c];
    }
}

// ---------------------------------------------------------------------------
// Launch: dict order = src, pos, reference_points, w_value, b_value, w_off,
// b_off, w_attn, b_attn, w_out, b_out, w_ff1, b_ff1, w_ff2, b_ff2,
// ln1_w, ln1_b, ln2_w, ln2_b. All fp32.
// ---------------------------------------------------------------------------
extern "C" void kernel_launch(void* const* d_in, const int* in_sizes, int n_in,
                              void* d_out, int out_size, void* d_ws, size_t ws_size,
                              hipStream_t stream) {
    const float* src    = (const float*)d_in[0];
    const float* pos    = (const float*)d_in[1];
    const float* refp   = (const float*)d_in[2];
    const float* w_val  = (const float*)d_in[3];
    const float* b_val  = (const float*)d_in[4];
    const float* w_off  = (const float*)d_in[5];
    const float* b_off  = (const float*)d_in[6];
    const float* w_attn = (const float*)d_in[7];
    const float* b_attn = (const float*)d_in[8];
    const float* w_out  = (const float*)d_in[9];
    const float* b_out  = (const float*)d_in[10];
    const float* w_ff1  = (const float*)d_in[11];
    const float* b_ff1  = (const float*)d_in[12];
    const float* w_ff2  = (const float*)d_in[13];
    const float* b_ff2  = (const float*)d_in[14];
    const float* ln1w   = (const float*)d_in[15];
    const float* ln1b   = (const float*)d_in[16];
    const float* ln2w   = (const float*)d_in[17];
    const float* ln2b   = (const float*)d_in[18];
    float* out = (float*)d_out;

    // Workspace layout (floats). Reuse: ca overwrites q (dead after the
    // off/attn projections); ff2 output overwrites value (dead after sampling).
    // Total = Mc * (5*256 + 128 + 1024) floats ~= 259 MB.
    float* ws   = (float*)d_ws;
    float* q    = ws;                        // Mc*256   (later: ca)
    float* val  = q    + (size_t)Mc * 256;   // Mc*256   (later: ff2 out)
    float* off  = val  + (size_t)Mc * 256;   // Mc*256
    float* attn = off  + (size_t)Mc * 256;   // Mc*128
    float* x    = attn + (size_t)Mc * 128;   // Mc*256
    float* acc  = x    + (size_t)Mc * 256;   // Mc*256
    float* ff   = acc  + (size_t)Mc * 256;   // Mc*1024

    const int  nelem = Mc * 256;
    const dim3 blk(256);
    const int  myt = (Mc + 63) / 64;   // 64-row block tiles
    const dim3 g256(2, myt);    // N = 256
    const dim3 g128(1, myt);    // N = 128
    const dim3 g1024(8, myt);   // N = 1024

    // 1) q = src + pos
    add_kernel<<<(nelem + 255) / 256, blk, 0, stream>>>(q, src, pos, nelem);

    // 2) value / offset / attention projections (WMMA fp32)
    gemm_wmma_kernel<0><<<g256, blk, 0, stream>>>(src, w_val, b_val, val, Mc, 256, 256);
    gemm_wmma_kernel<0><<<g256, blk, 0, stream>>>(q, w_off, b_off, off, Mc, 256, 256);
    gemm_wmma_kernel<0><<<g128, blk, 0, stream>>>(q, w_attn, b_attn, attn, Mc, 128, 256);

    // 3) softmax over the 16 sampling weights per (b,s,head)
    const int nrows16 = Mc * NHc;
    softmax16_kernel<<<(nrows16 + 255) / 256, blk, 0, stream>>>(attn, nrows16);

    // 4) deformable bilinear sampling + attention weighting
    const int nwaves = Mc * NHc;          // one wave32 per (b,s,head)
    deform_attn_kernel<<<(nwaves + 7) / 8, blk, 0, stream>>>(val, off, attn, refp, acc);

    // 5) output projection, residual + LN1
    gemm_wmma_kernel<0><<<g256, blk, 0, stream>>>(acc, w_out, b_out, q /*ca*/, Mc, 256, 256);
    residual_ln_kernel<<<(Mc + 7) / 8, blk, 0, stream>>>(src, q, ln1w, ln1b, x, Mc);

    // 6) FFN: relu(x@w_ff1+b) @ w_ff2 + b, residual + LN2 -> out
    gemm_wmma_kernel<1><<<g1024, blk, 0, stream>>>(x, w_ff1, b_ff1, ff, Mc, 1024, 256);
    gemm_wmma_kernel<0><<<g256, blk, 0, stream>>>(ff, w_ff2, b_ff2, val /*ff2o*/, Mc, 256, 1024);
    residual_ln_kernel<<<(Mc + 7) / 8, blk, 0, stream>>>(x, val, ln2w, ln2b, out, Mc);
}